// ModelNew_66924180406834
// MI455X (gfx1250) — compile-verified
//
#include <hip/hip_runtime.h>
#include <math.h>

// Problem constants (from reference)
constexpr int BATCH = 128;
constexpr int CIN   = 3;
constexpr int HH    = 256;
constexpr int WW    = 256;
constexpr int COUT  = 16;
constexpr int KK    = 3;

constexpr int PLANE  = HH * WW;          // 65536 floats per (b,ci) plane
constexpr int CHUNKS = 8;                // blocks per plane
constexpr int CHUNK  = PLANE / CHUNKS;   // 8192 floats per block
constexpr int NPLANE = BATCH * CIN;      // 384 planes
constexpr int NPART  = NPLANE * CHUNKS;  // 3072 partials

typedef __attribute__((ext_vector_type(2))) float v2f;
typedef __attribute__((ext_vector_type(8))) float v8f;

// ---------------------------------------------------------------------------
// Kernel 1: HBM-bound reduction. Each block sums an 8192-float chunk of one
// (b,ci) plane via float4 (global_load_b128) loads and writes one partial.
// 3072 blocks x 256 threads; each thread sums 8 x float4 = 32 floats.
// Deterministic LDS tree reduction (no atomics -> graph-replay stable).
// ---------------------------------------------------------------------------
__global__ void plane_chunk_sum(const float* __restrict__ x,
                                float* __restrict__ partial) {
    const int bid   = blockIdx.x;       // 0..NPART-1
    const int plane = bid >> 3;         // / CHUNKS
    const int chunk = bid & (CHUNKS - 1);
    const int tid   = threadIdx.x;      // 0..255

    const float4* p = reinterpret_cast<const float4*>(
        x + (size_t)plane * PLANE + (size_t)chunk * CHUNK);

    float s = 0.0f;
#pragma unroll
    for (int i = 0; i < 8; ++i) {
        float4 v = p[tid + i * 256];    // 2048 float4 per chunk, fully coalesced
        s += (v.x + v.y) + (v.z + v.w);
    }

    __shared__ float red[256];
    red[tid] = s;
    __syncthreads();
#pragma unroll
    for (int off = 128; off > 0; off >>= 1) {
        if (tid < off) red[tid] += red[tid + off];
        __syncthreads();
    }
    if (tid == 0) partial[bid] = red[0];
}

// ---------------------------------------------------------------------------
// Kernel 2: single block, 256 threads (8 waves of 32).
//  - fold 3072 partials -> S[b][ci] in LDS (ci padded to 4 with 0)
//  - fold weights -> Wsum[ci][co] * 1/((H+2)(W+2)) in LDS (ci=3 row zero)
//  - 8 batch tiles of 16: one V_WMMA_F32_16X16X4_F32 per wave computes
//    Y[16x16] = S_tile[16x4] x Wsum[4x16]
//  - add bias, logsumexp over the 16 channels per batch, scale by 10
// ---------------------------------------------------------------------------
__global__ void wmma_finalize(const float* __restrict__ partial,
                              const float* __restrict__ weight,  // [CIN][COUT][3][3]
                              const float* __restrict__ bias,    // [COUT]
                              float* __restrict__ out) {         // [BATCH]
    __shared__ float sS[BATCH][4];     // S[b][ci], ci=3 is zero pad
    __shared__ float sW[4][COUT];      // Wsum[ci][co] (pre-scaled), ci=3 zero
    __shared__ float sBias[COUT];
    __shared__ float sY[8][16][16];    // per-tile 16x16 WMMA results

    const int tid = threadIdx.x;
    const float invA = 1.0f / (float)((HH + KK - 1) * (WW + KK - 1)); // 1/258^2

    if (tid < BATCH) sS[tid][3] = 0.0f;
    if (tid < COUT) { sW[3][tid] = 0.0f; sBias[tid] = bias[tid]; }

    // S[b][ci] = sum of 8 chunk partials of plane p = b*3+ci
    for (int p = tid; p < NPLANE; p += 256) {
        float s = 0.0f;
#pragma unroll
        for (int c = 0; c < CHUNKS; ++c) s += partial[p * CHUNKS + c];
        sS[p / CIN][p % CIN] = s;
    }

    // Wsum[ci][co] = (sum of 3x3 kernel) * invA
    if (tid < CIN * COUT) {
        const int ci = tid / COUT, co = tid % COUT;
        const float* wp = weight + (size_t)(ci * COUT + co) * (KK * KK);
        float s = 0.0f;
#pragma unroll
        for (int i = 0; i < KK * KK; ++i) s += wp[i];
        sW[ci][co] = s * invA;
    }
    __syncthreads();

    // --- WMMA: one 16(batch) x 16(cout) x 4(cin) f32 MMA per wave ---
    const int wave = tid >> 5;
    const int lane = tid & 31;
    const int half = lane >> 4;   // 0: K=0,1 ; 1: K=2,3
    const int l16  = lane & 15;
    const int m    = wave * 16 + l16;   // batch row of this lane

    // A (16x4): VGPR0 = K=0 (lanes 0-15) / K=2 (lanes 16-31); VGPR1 = K=1 / K=3
    v2f a;
    a.x = half ? sS[m][2] : sS[m][0];
    a.y = half ? sS[m][3] : sS[m][1];   // K=3 pad = 0 for upper half
    // B (4x16): VGPR0 = K=0 / K=2 rows; VGPR1 = K=1 / K=3; N = l16 across lanes
    v2f b;
    b.x = half ? sW[2][l16] : sW[0][l16];
    b.y = half ? sW[3][l16] : sW[1][l16];

    v8f c = {};
    c = __builtin_amdgcn_wmma_f32_16x16x4_f32(
        /*neg_a=*/false, a, /*neg_b=*/false, b,
        /*c_mod=*/(short)0, c, /*reuse_a=*/false, /*reuse_b=*/false);

    // D layout: vgpr i, lane l -> (M = i + 8*(l>=16), N = l&15)
    const float bb = sBias[l16];
#pragma unroll
    for (int i = 0; i < 8; ++i) {
        sY[wave][i + half * 8][l16] = c[i] + bb;
    }
    __syncthreads();

    // logsumexp over the 16 channels, one thread per batch element
    if (tid < BATCH) {
        const float* row = sY[tid >> 4][tid & 15];
        float mx = row[0];
#pragma unroll
        for (int n = 1; n < 16; ++n) mx = fmaxf(mx, row[n]);
        float se = 0.0f;
#pragma unroll
        for (int n = 0; n < 16; ++n) se += expf(row[n] - mx);
        out[tid] = 10.0f * (mx + logf(se));
    }
}

// ---------------------------------------------------------------------------
// Launch: inputs are {x[128*3*256*256] f32, weight[3*16*3*3] f32, bias[16] f32}
// output: 128 f32. d_ws holds 3072 partial floats (12 KB), rewritten fully
// every call (no reliance on prior state).
// ---------------------------------------------------------------------------
extern "C" void kernel_launch(void* const* d_in, const int* in_sizes, int n_in,
                              void* d_out, int out_size, void* d_ws, size_t ws_size,
                              hipStream_t stream) {
    const float* x      = (const float*)d_in[0];
    const float* weight = (const float*)d_in[1];
    const float* bias   = (const float*)d_in[2];
    float* out          = (float*)d_out;
    float* partial      = (float*)d_ws;   // NPART floats

    plane_chunk_sum<<<NPART, 256, 0, stream>>>(x, partial);
    wmma_finalize<<<1, 256, 0, stream>>>(partial, weight, bias, out);
}